// GCNLayer_10771777979054
// MI455X (gfx1250) — compile-verified
//
#include <hip/hip_runtime.h>

// ---------------------------------------------------------------------------
// GCN layer for MI455X (gfx1250, wave32):
//   h1 = D_norm * scatter_sum(features[src] * edge_w, dst)
//   h2 = D_norm * scatter_sum(h1[src]       * edge_w, dst)
//   out = concat(relu(h0@W0+b0), relu(h1@W1+b1), relu(h2@W2+b2))  [10000,1536]
// Dense part uses V_WMMA_F32_16X16X4_F32 (exact f32, matches reference math).
// ---------------------------------------------------------------------------

typedef __attribute__((ext_vector_type(2))) float v2f;
typedef __attribute__((ext_vector_type(8))) float v8f;

#define N_NODES 10000
#define N_EDGES 160000
#define IN_DIM  512
#define OUT_DIM 512
#define LDS_STRIDE 516   // 516 mod 64 == 4 -> 16 rows land on distinct banks

// ---------------------------------------------------------------- zero buffer
__global__ void zero_f32(float4* __restrict__ p) {
    int gid = blockIdx.x * blockDim.x + threadIdx.x;   // exact-sized grid
    p[gid] = make_float4(0.f, 0.f, 0.f, 0.f);
}

// ------------------------------------------------- SpMM: gather-scale-scatter
// One float4 chunk (4 features) per thread; 128 chunks per edge.
__global__ void spmm_scatter(const float* __restrict__ h,
                             const float* __restrict__ ew,
                             const int*   __restrict__ src,
                             const int*   __restrict__ dst,
                             float*       __restrict__ agg) {
    int gid = blockIdx.x * blockDim.x + threadIdx.x;
    int e = gid >> 7;          // edge index
    int q = gid & 127;         // float4 chunk within the 512-wide row
    if (e >= N_EDGES) return;
    int s = src[e];
    int d = dst[e];
    float w = ew[e];
    const float4* hs = (const float4*)(h + (size_t)s * IN_DIM);
    float4 v = hs[q];
    float* o = agg + (size_t)d * IN_DIM + q * 4;
    atomicAdd(o + 0, v.x * w);
    atomicAdd(o + 1, v.y * w);
    atomicAdd(o + 2, v.z * w);
    atomicAdd(o + 3, v.w * w);
}

// ------------------------------------------------------- per-row D_norm scale
__global__ void scale_rows(float4* __restrict__ h, const float* __restrict__ dn) {
    int gid = blockIdx.x * blockDim.x + threadIdx.x;   // one float4 each
    int row = gid >> 7;                                // 128 float4 per row
    float s = dn[row];
    float4 v = h[gid];
    v.x *= s; v.y *= s; v.z *= s; v.w *= s;
    h[gid] = v;
}

// --------------------------------------------------- GEMM + bias + ReLU (f32)
// Block: 256 threads = 8 waves. Block tile: 16 rows x 128 cols.
// Wave w handles cols [bn*128 + w*16, +16). A-tile (16x512) staged in LDS.
// WMMA f32 16x16x4 layouts (wave32):
//   A (16x4):  lane holds row = lane&15; k-pair base = 2*(lane>>4); 2 VGPRs
//   B (4x16):  lane holds col = lane&15; k rows kb, kb+1 with kb = 2*(lane>>4)
//   C/D:       lane l, vgpr v -> (row = v + 8*(l>>4), col = l&15)
__global__ __launch_bounds__(256)
void gemm_relu_wmma(const float* __restrict__ A,     // [M x 512]
                    const float* __restrict__ W,     // [512 x 512] (in x out)
                    const float* __restrict__ bias,  // [512]
                    float*       __restrict__ out,   // base already col-offset
                    int ldo) {
    __shared__ float As[16 * LDS_STRIDE];            // ~32.3 KB of 320 KB/WGP

    const int tid  = threadIdx.x;
    const int wave = tid >> 5;
    const int lane = tid & 31;

    const int row0 = blockIdx.x * 16;                // 625 blocks, M=10000 exact
    const int col0 = blockIdx.y * 128 + wave * 16;

    // ---- stage A tile 16x512 (coalesced float4, 8 per thread) ----
    {
        const float4* Ag = (const float4*)(A + (size_t)row0 * IN_DIM);
#pragma unroll
        for (int i = 0; i < 8; ++i) {
            int idx = tid + i * 256;                 // 0..2047
            int r   = idx >> 7;                      // row 0..15
            int c4  = idx & 127;                     // float4 col
            float4 v = Ag[(size_t)r * 128 + c4];
            float* dp = &As[r * LDS_STRIDE + c4 * 4];
            dp[0] = v.x; dp[1] = v.y; dp[2] = v.z; dp[3] = v.w;
        }
    }
    __syncthreads();

    const int arow = lane & 15;
    const int kgrp = (lane >> 4) * 2;                // 0 or 2
    const int bcol = col0 + (lane & 15);
    const float* Wc = W + bcol;

    v8f acc = {};
#pragma unroll 4
    for (int k4 = 0; k4 < IN_DIM / 4; ++k4) {
        int kb = k4 * 4 + kgrp;
        v2f a = *(const v2f*)&As[arow * LDS_STRIDE + kb];   // 8B-aligned LDS read
        v2f b;
        b.x = Wc[(size_t)kb * OUT_DIM];
        b.y = Wc[(size_t)(kb + 1) * OUT_DIM];
        acc = __builtin_amdgcn_wmma_f32_16x16x4_f32(
            /*neg_a=*/false, a, /*neg_b=*/false, b,
            /*c_mod=*/(short)0, acc, /*reuse_a=*/false, /*reuse_b=*/false);
    }

    const float bb = bias[bcol];
#pragma unroll
    for (int v = 0; v < 8; ++v) {
        int r = row0 + v + 8 * (lane >> 4);
        float val = acc[v] + bb;
        val = val > 0.f ? val : 0.f;
        out[(size_t)r * ldo + bcol] = val;
    }
}

// ---------------------------------------------------------------------------
extern "C" void kernel_launch(void* const* d_in, const int* in_sizes, int n_in,
                              void* d_out, int out_size, void* d_ws, size_t ws_size,
                              hipStream_t stream) {
    const float* features = (const float*)d_in[0];
    const float* D_norm   = (const float*)d_in[1];
    const float* edge_w   = (const float*)d_in[2];
    const float* W        = (const float*)d_in[3];
    const float* b        = (const float*)d_in[4];
    const int*   src      = (const int*)d_in[5];
    const int*   dst      = (const int*)d_in[6];
    float*       out      = (float*)d_out;

    // workspace: two hop buffers, 10000*512 f32 each (~41 MB total)
    float* h1 = (float*)d_ws;
    float* h2 = h1 + (size_t)N_NODES * IN_DIM;

    const int vec4   = (N_NODES * IN_DIM) / 4;       // 1,280,000 float4
    const int zgrid  = vec4 / 256;                   // 5000 (exact)
    const int egrid  = (N_EDGES * 128 + 255) / 256;  // 80000

    // hop 1
    zero_f32   <<<zgrid, 256, 0, stream>>>((float4*)h1);
    spmm_scatter<<<egrid, 256, 0, stream>>>(features, edge_w, src, dst, h1);
    scale_rows <<<zgrid, 256, 0, stream>>>((float4*)h1, D_norm);
    // hop 2
    zero_f32   <<<zgrid, 256, 0, stream>>>((float4*)h2);
    spmm_scatter<<<egrid, 256, 0, stream>>>(h1, edge_w, src, dst, h2);
    scale_rows <<<zgrid, 256, 0, stream>>>((float4*)h2, D_norm);

    // dense heads: out[:, hop*512 : (hop+1)*512] = relu(h @ W[hop] + b[hop])
    dim3 ggrid(N_NODES / 16, OUT_DIM / 128);         // (625, 4)
    const int ldo = 3 * OUT_DIM;                     // 1536
    gemm_relu_wmma<<<ggrid, 256, 0, stream>>>(
        features, W + 0 * IN_DIM * OUT_DIM, b + 0 * OUT_DIM, out + 0 * OUT_DIM, ldo);
    gemm_relu_wmma<<<ggrid, 256, 0, stream>>>(
        h1,       W + 1 * IN_DIM * OUT_DIM, b + 1 * OUT_DIM, out + 1 * OUT_DIM, ldo);
    gemm_relu_wmma<<<ggrid, 256, 0, stream>>>(
        h2,       W + 2 * IN_DIM * OUT_DIM, b + 2 * OUT_DIM, out + 2 * OUT_DIM, ldo);
}